// Epipolar_17901423689882
// MI455X (gfx1250) — compile-verified
//
#include <hip/hip_runtime.h>
#include <math.h>
#include <stdint.h>

// Problem constants (fixed by setup_inputs): N=2, C=256, H=W=64, S=64
#define NN   2
#define CC   256
#define HH   64
#define WW   64
#define SS   64
#define CB   16          // channels staged per LDS tile
#define ROWF 20          // floats per pixel row in LDS (16 data + 4 pad) -> 320KB, 16B-aligned rows
#define TPB  512         // 16 waves per workgroup (max residency at 320KB/WGP)
#define EPSL 1e-3f

typedef float v4f __attribute__((ext_vector_type(4)));

__device__ __forceinline__ v4f splat4(float x) { v4f v = {x, x, x, x}; return v; }

// ---- CDNA5 async global->LDS scatter (per-lane LDS dest address) ----------
__device__ __forceinline__ void async_ld_b32(uint32_t lds_byte_off, const float* gaddr) {
    asm volatile("global_load_async_to_lds_b32 %0, %1, off"
                 :: "v"(lds_byte_off),
                    "v"((unsigned long long)(uintptr_t)gaddr)
                 : "memory");
}

__device__ __forceinline__ void wait_async0() {
#if __has_builtin(__builtin_amdgcn_s_wait_asynccnt)
    __builtin_amdgcn_s_wait_asynccnt(0);
#else
    asm volatile("s_wait_asynccnt 0" ::: "memory");
#endif
}

// ---- fundamental matrix from P1 (q, 3x4 row-major) and P2 (r, 3x4) --------
__device__ __forceinline__ void compute_F(const float* __restrict__ q,
                                          const float* __restrict__ r,
                                          float F[9]) {
    auto d3 = [q](int a, int b, int c) {
        float m00 = q[a],     m01 = q[b],     m02 = q[c];
        float m10 = q[4 + a], m11 = q[4 + b], m12 = q[4 + c];
        float m20 = q[8 + a], m21 = q[8 + b], m22 = q[8 + c];
        return m00 * (m11 * m22 - m12 * m21)
             - m01 * (m10 * m22 - m12 * m20)
             + m02 * (m10 * m21 - m11 * m20);
    };
    // Null vector of q == camera center == smallest-eigenvector of qT q
    float C0 =  d3(1, 2, 3);
    float C1 = -d3(0, 2, 3);
    float C2 =  d3(0, 1, 3);
    float C3 = -d3(0, 1, 2);
    float inv = rsqrtf(C0 * C0 + C1 * C1 + C2 * C2 + C3 * C3);
    C0 *= inv; C1 *= inv; C2 *= inv; C3 *= inv;

    // e2 = P2 @ Cc
    float ex_ = r[0] * C0 + r[1] * C1 + r[2]  * C2 + r[3]  * C3;
    float ey_ = r[4] * C0 + r[5] * C1 + r[6]  * C2 + r[7]  * C3;
    float ez_ = r[8] * C0 + r[9] * C1 + r[10] * C2 + r[11] * C3;

    // A = q qT (3x3 symmetric), Ainv via adjugate
    float A00 = 0.f, A01 = 0.f, A02 = 0.f, A11 = 0.f, A12 = 0.f, A22 = 0.f;
#pragma unroll
    for (int k = 0; k < 4; ++k) {
        A00 += q[k] * q[k];         A01 += q[k] * q[4 + k];     A02 += q[k] * q[8 + k];
        A11 += q[4 + k] * q[4 + k]; A12 += q[4 + k] * q[8 + k]; A22 += q[8 + k] * q[8 + k];
    }
    float c00 =  (A11 * A22 - A12 * A12);
    float c01 = -(A01 * A22 - A12 * A02);
    float c02 =  (A01 * A12 - A11 * A02);
    float det = A00 * c00 + A01 * c01 + A02 * c02;
    float id  = 1.f / det;
    float i00 = c00 * id, i01 = c01 * id, i02 = c02 * id;
    float i11 = (A00 * A22 - A02 * A02) * id;
    float i12 = -(A00 * A12 - A01 * A02) * id;
    float i22 = (A00 * A11 - A01 * A01) * id;
    float AI[3][3] = {{i00, i01, i02}, {i01, i11, i12}, {i02, i12, i22}};

    // pinv(P1) = P1^T Ainv  (4x3), then B = P2 @ pinv  (3x3)
    float PV[4][3];
#pragma unroll
    for (int k = 0; k < 4; ++k)
#pragma unroll
        for (int j = 0; j < 3; ++j)
            PV[k][j] = q[k] * AI[0][j] + q[4 + k] * AI[1][j] + q[8 + k] * AI[2][j];
    float B[3][3];
#pragma unroll
    for (int i = 0; i < 3; ++i)
#pragma unroll
        for (int j = 0; j < 3; ++j)
            B[i][j] = r[i * 4 + 0] * PV[0][j] + r[i * 4 + 1] * PV[1][j]
                    + r[i * 4 + 2] * PV[2][j] + r[i * 4 + 3] * PV[3][j];

    // F = skew(e2) @ B
    F[0] = -ez_ * B[1][0] + ey_ * B[2][0];
    F[1] = -ez_ * B[1][1] + ey_ * B[2][1];
    F[2] = -ez_ * B[1][2] + ey_ * B[2][2];
    F[3] =  ez_ * B[0][0] - ex_ * B[2][0];
    F[4] =  ez_ * B[0][1] - ex_ * B[2][1];
    F[5] =  ez_ * B[0][2] - ex_ * B[2][2];
    F[6] = -ey_ * B[0][0] + ex_ * B[1][0];
    F[7] = -ey_ * B[0][1] + ex_ * B[1][1];
    F[8] = -ey_ * B[0][2] + ex_ * B[1][2];
}

extern __shared__ float tile[];   // HH*WW*ROWF floats = 320 KB, layout [pixel][c(16)+pad(4)]

__global__ void __launch_bounds__(TPB)
epipolar_sample_max_kernel(const float* __restrict__ feat2,
                           const float* __restrict__ P1,
                           const float* __restrict__ P2,
                           float* __restrict__ out) {
    const int tid = threadIdx.x;
    const int pb  = blockIdx.x;   // pixel block  0..7   (512 pixels each)
    const int cb  = blockIdx.y;   // channel tile 0..15  (16 channels each)
    const int n   = blockIdx.z;   // batch        0..1

    // -------- Stage 16 channel planes into LDS, channel-interleaved --------
    // Coalesced global reads; the async per-lane LDS scatter performs the
    // planar (C,H,W) -> interleaved (H,W,C16) transpose for free (ASYNCcnt).
    {
        const float* g0 = feat2 + (size_t)(n * CC + cb * CB) * (HH * WW) + tid;
        const uint32_t l0 = (uint32_t)(tid * ROWF) * 4u;
#pragma unroll
        for (int c = 0; c < CB; ++c) {
#pragma unroll
            for (int k = 0; k < (HH * WW) / TPB; ++k) {
                async_ld_b32(l0 + (uint32_t)(k * TPB * ROWF + c) * 4u,
                             g0 + c * (HH * WW) + k * TPB);
            }
        }
    }
    wait_async0();
    __syncthreads();

    // -------- Per-pixel epipolar line geometry --------
    float F[9];
    compute_F(P1 + n * 12, P2 + n * 12, F);

    const int p  = pb * TPB + tid;     // output pixel
    const float wf = (float)(p & 63);  // x
    const float hf = (float)(p >> 6);  // y

    float a_ = F[0] * wf + F[1] * hf + F[2];
    float b_ = F[3] * wf + F[4] * hf + F[5];
    float c_ = F[6] * wf + F[7] * hf + F[8];
    float db = copysignf(fmaxf(fabsf(b_), EPSL), b_);
    float da = copysignf(fmaxf(fabsf(a_), EPSL), a_);
    float by1 = -c_ / db;
    float by2 = -(63.f * a_ + c_) / db;
    float bx0 = -c_ / da;
    float bx3 = -(63.f * b_ + c_) / da;

    bool m0 = (bx0 >= EPSL) && (bx0 <  63.f - EPSL);
    bool m1 = (by1 >  EPSL) && (by1 <= 63.f - EPSL);
    bool m2 = (by2 >= EPSL) && (by2 <  63.f - EPSL);
    bool m3 = (bx3 >  EPSL) && (bx3 <= 63.f - EPSL);
    int  nint   = (int)m0 + (int)m1 + (int)m2 + (int)m3;
    bool valid2 = (nint >= 2);
    if (!valid2) { m0 = true; m1 = true; m2 = false; m3 = false; }

    // first two valid intersections in order p0,p1,p2,p3 (== reference argsort)
    float sxx, syy, exx, eyy;
    if (m0) {
        sxx = bx0; syy = 0.f;
        if (m1)      { exx = 0.f;  eyy = by1; }
        else if (m2) { exx = 63.f; eyy = by2; }
        else         { exx = bx3;  eyy = 63.f; }
    } else if (m1) {
        sxx = 0.f; syy = by1;
        if (m2)      { exx = 63.f; eyy = by2; }
        else         { exx = bx3;  eyy = 63.f; }
    } else {
        sxx = 63.f; syy = by2; exx = bx3; eyy = 63.f;
    }
    if (!valid2) { sxx = syy = exx = eyy = -10000.f; }   // -> all weights 0 -> out 0

    // locs -> pixel coords: x = lx*(64/63) - 0.5, stepping lx by vec/63
    const float K  = 64.f / 63.f;
    float xb = fmaf(sxx, K, -0.5f);
    float yb = fmaf(syy, K, -0.5f);
    float dx = (exx - sxx) * (K / 63.f);
    float dy = (eyy - syy) * (K / 63.f);

    // -------- 64 steps x 4 corners x 16 channels, max-accumulate --------
    v4f acc[4];
#pragma unroll
    for (int g = 0; g < 4; ++g)
        acc[g] = splat4(-INFINITY);

#pragma unroll 2
    for (int s = 0; s < SS; ++s) {
        float x  = fmaf((float)s, dx, xb);
        float y  = fmaf((float)s, dy, yb);
        float xf = floorf(x), yf = floorf(y);
        float fx = x - xf,    fy = y - yf;
        int ix0 = (int)xf, iy0 = (int)yf;
        int ix1 = ix0 + 1,  iy1 = iy0 + 1;

        float w00 = (1.f - fx) * (1.f - fy);
        float w10 = fx * (1.f - fy);
        float w01 = (1.f - fx) * fy;
        float w11 = fx * fy;
        bool vx0 = ((unsigned)ix0 < 64u), vx1 = ((unsigned)ix1 < 64u);
        bool vy0 = ((unsigned)iy0 < 64u), vy1 = ((unsigned)iy1 < 64u);
        w00 = (vx0 && vy0) ? w00 : 0.f;
        w10 = (vx1 && vy0) ? w10 : 0.f;
        w01 = (vx0 && vy1) ? w01 : 0.f;
        w11 = (vx1 && vy1) ? w11 : 0.f;

        int cx0 = ix0 < 0 ? 0 : (ix0 > 63 ? 63 : ix0);   // -> v_med3_i32
        int cx1 = ix1 < 0 ? 0 : (ix1 > 63 ? 63 : ix1);
        int cy0 = iy0 < 0 ? 0 : (iy0 > 63 ? 63 : iy0);
        int cy1 = iy1 < 0 ? 0 : (iy1 > 63 ? 63 : iy1);

        const float* t00 = tile + (cy0 * 64 + cx0) * ROWF;
        const float* t10 = tile + (cy0 * 64 + cx1) * ROWF;
        const float* t01 = tile + (cy1 * 64 + cx0) * ROWF;
        const float* t11 = tile + (cy1 * 64 + cx1) * ROWF;

        const v4f W00 = splat4(w00), W10 = splat4(w10);
        const v4f W01 = splat4(w01), W11 = splat4(w11);

#pragma unroll
        for (int g = 0; g < 4; ++g) {
            v4f va = *(const v4f*)(t00 + 4 * g);   // ds_load_b128: 4 channels
            v4f vb = *(const v4f*)(t10 + 4 * g);
            v4f vc = *(const v4f*)(t01 + 4 * g);
            v4f vd = *(const v4f*)(t11 + 4 * g);
            // packed-friendly: backend can select v_pk_fma_f32 / VOPD pairs
            v4f r = __builtin_elementwise_fma(W11, vd,
                    __builtin_elementwise_fma(W01, vc,
                    __builtin_elementwise_fma(W10, vb, W00 * va)));
            acc[g] = __builtin_elementwise_max(acc[g], r);
        }
    }

    // -------- coalesced channel-planar stores --------
    float* dst = out + (size_t)(n * CC + cb * CB) * (HH * WW) + p;
#pragma unroll
    for (int g = 0; g < 4; ++g) {
        dst[(4 * g + 0) * (HH * WW)] = acc[g].x;
        dst[(4 * g + 1) * (HH * WW)] = acc[g].y;
        dst[(4 * g + 2) * (HH * WW)] = acc[g].z;
        dst[(4 * g + 3) * (HH * WW)] = acc[g].w;
    }
}

extern "C" void kernel_launch(void* const* d_in, const int* in_sizes, int n_in,
                              void* d_out, int out_size, void* d_ws, size_t ws_size,
                              hipStream_t stream) {
    (void)in_sizes; (void)n_in; (void)out_size; (void)d_ws; (void)ws_size;
    const float* feat2 = (const float*)d_in[1];   // feat1 (d_in[0]) is unused by the reference
    const float* P1    = (const float*)d_in[2];
    const float* P2    = (const float*)d_in[3];
    float* out = (float*)d_out;

    dim3 grid((HH * WW) / TPB, CC / CB, NN);   // (8 pixel blocks, 16 channel tiles, 2)
    dim3 block(TPB);
    size_t lds_bytes = (size_t)HH * WW * ROWF * sizeof(float);  // 320 KB
    epipolar_sample_max_kernel<<<grid, block, lds_bytes, stream>>>(feat2, P1, P2, out);
}